// GCNLayer_84181359002199
// MI455X (gfx1250) — compile-verified
//
#include <hip/hip_runtime.h>
#include <stdint.h>

typedef float v2f __attribute__((ext_vector_type(2)));
typedef float v8f __attribute__((ext_vector_type(8)));

#define FDIM 128      // IN == OUT == 128
#define APITCH 132    // LDS pitch (dwords): bank = col + 4*row (mod 64) -> conflict-free columns
#define MROWS 80      // rows per block: 100000 = 1250 * 80

__device__ __forceinline__ void atomAddF(float* p, float v) {
    __hip_atomic_fetch_add(p, v, __ATOMIC_RELAXED, __HIP_MEMORY_SCOPE_AGENT);
}

// ---------------- init: zero out, deg = 1.0 (self-loop) ----------------
__global__ void init_k(float* __restrict__ out, float* __restrict__ deg,
                       int NF, int N_) {
    int i = blockIdx.x * blockDim.x + threadIdx.x;
    if (i < NF) out[i] = 0.0f;
    if (i < N_) deg[i] = 1.0f;
}

// ---------------- degree over destinations ----------------
__global__ void degree_k(const long long* __restrict__ ei,
                         float* __restrict__ deg, int E_) {
    int e = blockIdx.x * blockDim.x + threadIdx.x;
    if (e >= E_) return;
    long long d = ei[(long long)E_ + e];   // edge_index[1][e]
    atomAddF(deg + d, 1.0f);
}

// ---------------- dis = deg^{-1/2} (deg >= 1 always) ----------------
__global__ void rsqrt_k(float* __restrict__ deg, int N_) {
    int v = blockIdx.x * blockDim.x + threadIdx.x;
    if (v < N_) deg[v] = rsqrtf(deg[v]);
}

// ---------------- h = x @ W via V_WMMA_F32_16X16X4_F32 ----------------
// Block: 256 threads = 8 waves, covers MROWS(=80) rows x 128 cols.
// Wave w owns columns [16w, 16w+16): preloads its whole 128x16 B-panel into
// 64 VGPRs (WMMA B layout), then 5 M-subtiles x 32 k-steps of pure LDS+WMMA.
__global__ __launch_bounds__(256) void gemm_wmma_k(
    const float* __restrict__ x, const float* __restrict__ Wm,
    float* __restrict__ h, int N_) {
    __shared__ float sA[MROWS * APITCH];
    const int m0 = blockIdx.x * MROWS;
    const int t  = threadIdx.x;

    // cooperative, coalesced 80x128 tile load (float4), zero-fill past N
    const int rows = (N_ - m0 < MROWS) ? (N_ - m0) : MROWS;
    for (int i = t; i < MROWS * (FDIM / 4); i += 256) {
        const int r  = i >> 5;        // 32 float4 per row
        const int c4 = (i & 31) * 4;
        float4 v = make_float4(0.f, 0.f, 0.f, 0.f);
        if (r < rows)
            v = *(const float4*)(x + (size_t)(m0 + r) * FDIM + c4);
        float* dst = &sA[r * APITCH + c4];
        dst[0] = v.x; dst[1] = v.y; dst[2] = v.z; dst[3] = v.w;
    }
    __syncthreads();

    const int lane   = t & 31;
    const int lane15 = lane & 15;
    const int sel    = lane >> 4;        // 0: K pair {0,1}; 1: K pair {2,3}
    const int n0     = (t >> 5) * 16;    // wave's column tile

    // B 4x16 (KxN) per k-step: lanes = N; lanes 16-31 hold K+2 rows.
    v2f bReg[32];
    #pragma unroll
    for (int ks = 0; ks < 32; ++ks) {
        const int kk = ks * 4 + 2 * sel;
        bReg[ks].x = Wm[(size_t)kk * FDIM + n0 + lane15];
        bReg[ks].y = Wm[(size_t)(kk + 1) * FDIM + n0 + lane15];
    }

    for (int mt = 0; mt < MROWS; mt += 16) {
        v8f acc = {};
        #pragma unroll
        for (int ks = 0; ks < 32; ++ks) {
            const int kk = ks * 4 + 2 * sel;
            v2f a;
            // A 16x4 (MxK): lanes 0-15 M rows, K={kk,kk+1}; lanes 16-31 K+2
            a.x = sA[(mt + lane15) * APITCH + kk];
            a.y = sA[(mt + lane15) * APITCH + kk + 1];
            acc = __builtin_amdgcn_wmma_f32_16x16x4_f32(
                      false, a, false, bReg[ks], (short)0, acc, false, false);
        }
        // D 16x16 f32: VGPR r -> M=r (lanes 0-15) / M=r+8 (lanes 16-31)
        const int col   = n0 + lane15;
        const int rbase = m0 + mt + sel * 8;
        #pragma unroll
        for (int r = 0; r < 8; ++r) {
            const int gr = rbase + r;
            if (gr < N_) h[(size_t)gr * FDIM + col] = acc[r];
        }
    }
}

// ---------------- edge scatter: one wave32 per edge ----------------
// Lane l handles features [4l, 4l+4): coalesced 512B gather of h[src],
// 4x fp32 global atomic adds into out[dst].
__global__ __launch_bounds__(256) void scatter_k(
    const long long* __restrict__ ei, const float* __restrict__ h,
    const float* __restrict__ dis, float* __restrict__ out, int E_) {
    const int widx = (int)((blockIdx.x * (unsigned)blockDim.x + threadIdx.x) >> 5);
    const int lane = threadIdx.x & 31;
    if (widx >= E_) return;
    const long long s = ei[widx];
    const long long d = ei[(long long)E_ + widx];
    const float w = dis[s] * dis[d];
    const float4 hv = *(const float4*)(h + (size_t)s * FDIM + lane * 4);
    float* op = out + (size_t)d * FDIM + lane * 4;
    atomAddF(op + 0, hv.x * w);
    atomAddF(op + 1, hv.y * w);
    atomAddF(op + 2, hv.z * w);
    atomAddF(op + 3, hv.w * w);
}

// ---------------- finalize: self-loop + bias + ReLU ----------------
__global__ void finalize_k(float* __restrict__ out, const float* __restrict__ h,
                           const float* __restrict__ dis,
                           const float* __restrict__ b, int NF) {
    int i = blockIdx.x * blockDim.x + threadIdx.x;
    if (i >= NF) return;
    const int node = i >> 7, col = i & 127;
    const float ds = dis[node];
    const float v = out[i] + h[i] * ds * ds + b[col];
    out[i] = v > 0.0f ? v : 0.0f;
}

extern "C" void kernel_launch(void* const* d_in, const int* in_sizes, int n_in,
                              void* d_out, int out_size, void* d_ws, size_t ws_size,
                              hipStream_t stream) {
    const float*     x   = (const float*)d_in[0];
    const long long* ei  = (const long long*)d_in[1];   // int64 edge_index [2,E]
    const float*     Wm  = (const float*)d_in[2];
    const float*     b   = (const float*)d_in[3];
    float*           out = (float*)d_out;

    const int N_ = in_sizes[0] / FDIM;   // 100000
    const int E_ = in_sizes[1] / 2;      // 3200000
    const int NF = N_ * FDIM;

    // ws layout: [deg/dis: N floats][h: N*128 floats], 1KB aligned
    float* deg = (float*)d_ws;
    size_t hOff = ((size_t)N_ * sizeof(float) + 1023) & ~(size_t)1023;
    float* h = (float*)((char*)d_ws + hOff);

    init_k     <<<(NF + 255) / 256, 256, 0, stream>>>(out, deg, NF, N_);
    degree_k   <<<(E_ + 255) / 256, 256, 0, stream>>>(ei, deg, E_);
    rsqrt_k    <<<(N_ + 255) / 256, 256, 0, stream>>>(deg, N_);
    gemm_wmma_k<<<(N_ + MROWS - 1) / MROWS, 256, 0, stream>>>(x, Wm, h, N_);
    scatter_k  <<<(E_ + 7) / 8, 256, 0, stream>>>(ei, h, deg, out, E_);
    finalize_k <<<(NF + 255) / 256, 256, 0, stream>>>(out, h, deg, b, NF);
}